// RG_LRU_71786083386185
// MI455X (gfx1250) — compile-verified
//
#include <hip/hip_runtime.h>
#include <hip/hip_bf16.h>
#include <math.h>

// Problem constants (match reference)
#define BB 4
#define TT 4096
#define CC 1024
#define NN (BB * TT)            // 16384 rows of fused [B*T, C] GEMM

// GEMM tiling
#define BLK_M 64
#define BLK_N 128
#define BLK_K 16
#define KT (CC / BLK_K)         // 64 K-tiles
#define LDS_STRIDE (BLK_K + 4)  // 20 floats: conflict-free fragment reads

// Chunked scan
#define SEG 32
#define SEGLEN (TT / SEG)       // 128 timesteps per segment

typedef __attribute__((ext_vector_type(2))) float v2f;
typedef __attribute__((ext_vector_type(8))) float v8f;
typedef __attribute__((ext_vector_type(4))) int   v4i;

// Probe CDNA5 async global->LDS path; fall back to load+ds_store if absent.
#if defined(__has_builtin)
#if __has_builtin(__builtin_amdgcn_global_load_async_to_lds_b128) && \
    __has_builtin(__builtin_amdgcn_s_wait_asynccnt)
#define USE_ASYNC_LDS 1
#endif
#endif

#define AS1 __attribute__((address_space(1)))
#define AS3 __attribute__((address_space(3)))
typedef AS1 v4i* gptr128;   // global 128-bit pointer (builtin src type)
typedef AS3 v4i* lptr128;   // LDS    128-bit pointer (builtin dst type)

// ---------------------------------------------------------------------------
// Kernel 1: fused gate GEMMs (fp32 WMMA) + gate math, double-buffered LDS.
// ---------------------------------------------------------------------------
__global__ __launch_bounds__(256)
void rg_lru_gates_wmma(const float* __restrict__ x,
                       const float* __restrict__ Wi, const float* __restrict__ bi,
                       const float* __restrict__ Wr, const float* __restrict__ br,
                       const float* __restrict__ cap_lambda,
                       float* __restrict__ at_out, float* __restrict__ gx_out)
{
    __shared__ float xs [2][BLK_M * LDS_STRIDE];
    __shared__ float wis[2][BLK_N * LDS_STRIDE];
    __shared__ float wrs[2][BLK_N * LDS_STRIDE];

    const int tid  = threadIdx.x;
    const int lane = tid & 31;          // wave32
    const int wave = tid >> 5;          // 0..7
    const int wm   = wave >> 2;         // 0..1  (M dir, 32 rows each)
    const int wn   = wave & 3;          // 0..3  (N dir, 32 cols each)
    const int half = lane >> 4;         // K half of A/B frag
    const int l16  = lane & 15;

    const int m_blk = blockIdx.x * BLK_M;
    const int n_blk = blockIdx.y * BLK_N;

    // Per-thread staging slice: one float4 of x tile, two of each W tile.
    const int ldr = tid >> 2;           // 0..63
    const int ldq = (tid & 3) << 2;     // 0,4,8,12

    v8f acc_i[2][2] = {};
    v8f acc_r[2][2] = {};

    auto stage = [&](int k0, int buf) {
#if USE_ASYNC_LDS
        __builtin_amdgcn_global_load_async_to_lds_b128(
            (gptr128)(x + (size_t)(m_blk + ldr) * CC + k0 + ldq),
            (lptr128)&xs[buf][ldr * LDS_STRIDE + ldq], 0, 0);
        __builtin_amdgcn_global_load_async_to_lds_b128(
            (gptr128)(Wi + (size_t)(n_blk + ldr) * CC + k0 + ldq),
            (lptr128)&wis[buf][ldr * LDS_STRIDE + ldq], 0, 0);
        __builtin_amdgcn_global_load_async_to_lds_b128(
            (gptr128)(Wi + (size_t)(n_blk + ldr + 64) * CC + k0 + ldq),
            (lptr128)&wis[buf][(ldr + 64) * LDS_STRIDE + ldq], 0, 0);
        __builtin_amdgcn_global_load_async_to_lds_b128(
            (gptr128)(Wr + (size_t)(n_blk + ldr) * CC + k0 + ldq),
            (lptr128)&wrs[buf][ldr * LDS_STRIDE + ldq], 0, 0);
        __builtin_amdgcn_global_load_async_to_lds_b128(
            (gptr128)(Wr + (size_t)(n_blk + ldr + 64) * CC + k0 + ldq),
            (lptr128)&wrs[buf][(ldr + 64) * LDS_STRIDE + ldq], 0, 0);
#else
        const float4 vx  = *(const float4*)(x  + (size_t)(m_blk + ldr) * CC + k0 + ldq);
        const float4 wi0 = *(const float4*)(Wi + (size_t)(n_blk + ldr) * CC + k0 + ldq);
        const float4 wi1 = *(const float4*)(Wi + (size_t)(n_blk + ldr + 64) * CC + k0 + ldq);
        const float4 wr0 = *(const float4*)(Wr + (size_t)(n_blk + ldr) * CC + k0 + ldq);
        const float4 wr1 = *(const float4*)(Wr + (size_t)(n_blk + ldr + 64) * CC + k0 + ldq);
        *(float4*)&xs [buf][ldr * LDS_STRIDE + ldq]        = vx;
        *(float4*)&wis[buf][ldr * LDS_STRIDE + ldq]        = wi0;
        *(float4*)&wis[buf][(ldr + 64) * LDS_STRIDE + ldq] = wi1;
        *(float4*)&wrs[buf][ldr * LDS_STRIDE + ldq]        = wr0;
        *(float4*)&wrs[buf][(ldr + 64) * LDS_STRIDE + ldq] = wr1;
#endif
    };

    stage(0, 0);

    for (int kt = 0; kt < KT; ++kt) {
        const int cur = kt & 1;
#if USE_ASYNC_LDS
        __builtin_amdgcn_s_wait_asynccnt(0);   // this wave's tile-kt DMA done
#endif
        __syncthreads();                       // all waves: tile kt visible,
                                               // tile kt-1 consumers finished
        if (kt + 1 < KT) stage((kt + 1) * BLK_K, cur ^ 1);

        #pragma unroll
        for (int kk = 0; kk < BLK_K; kk += 4) {
            v2f a[2], bI[2], bR[2];
            #pragma unroll
            for (int mi = 0; mi < 2; ++mi) {
                // A 16x4 frag: lanes 0-15 -> K=0,1 ; lanes 16-31 -> K=2,3
                const float* p =
                    &xs[cur][(wm * 32 + mi * 16 + l16) * LDS_STRIDE + kk + 2 * half];
                a[mi] = (v2f){p[0], p[1]};
            }
            #pragma unroll
            for (int ni = 0; ni < 2; ++ni) {
                const float* pi =
                    &wis[cur][(wn * 32 + ni * 16 + l16) * LDS_STRIDE + kk + 2 * half];
                bI[ni] = (v2f){pi[0], pi[1]};
                const float* pr =
                    &wrs[cur][(wn * 32 + ni * 16 + l16) * LDS_STRIDE + kk + 2 * half];
                bR[ni] = (v2f){pr[0], pr[1]};
            }
            #pragma unroll
            for (int mi = 0; mi < 2; ++mi)
                #pragma unroll
                for (int ni = 0; ni < 2; ++ni) {
                    acc_i[mi][ni] = __builtin_amdgcn_wmma_f32_16x16x4_f32(
                        false, a[mi], false, bI[ni], (short)0, acc_i[mi][ni], false, false);
                    acc_r[mi][ni] = __builtin_amdgcn_wmma_f32_16x16x4_f32(
                        false, a[mi], false, bR[ni], (short)0, acc_r[mi][ni], false, false);
                }
        }
    }

    // ---- fused epilogue ----
    const float lam  = cap_lambda[0];
    const float coef = -8.0f * log1pf(__expf(lam));   // -8 * softplus(lambda)

    #pragma unroll
    for (int mi = 0; mi < 2; ++mi)
        #pragma unroll
        for (int ni = 0; ni < 2; ++ni) {
            const int col = n_blk + wn * 32 + ni * 16 + l16;   // N = lane % 16
            const float bii = bi[col];
            const float brr = br[col];
            #pragma unroll
            for (int v = 0; v < 8; ++v) {
                const int row = m_blk + wm * 32 + mi * 16 + v + 8 * half;
                const size_t idx = (size_t)row * CC + col;
                const float zi = acc_i[mi][ni][v] + bii;
                const float zr = acc_r[mi][ni][v] + brr;
                const float it = 1.0f / (1.0f + __expf(-zi));
                const float rt = 1.0f / (1.0f + __expf(-zr));
                const float at = __expf(coef * rt);
                const float g  = sqrtf(fmaxf(1.0f - at * at, 0.0f)) * it * x[idx];
                at_out[idx] = at;
                gx_out[idx] = g;
            }
        }
}

// ---------------------------------------------------------------------------
// Chunked scan: pass 1 — per-segment carry (P = prod a, G = scan from 0).
// One thread per (b,c,segment); consecutive threads -> consecutive c.
// ---------------------------------------------------------------------------
__global__ __launch_bounds__(256)
void rg_lru_scan_seg(const float* __restrict__ at, const float* __restrict__ gx,
                     float* __restrict__ segP, float* __restrict__ segG)
{
    const int tid = blockIdx.x * blockDim.x + threadIdx.x;   // 0 .. B*C*SEG-1
    const int s   = tid / (BB * CC);
    const int ch  = tid % (BB * CC);
    const int b   = ch / CC;
    const int c   = ch % CC;
    const size_t base = ((size_t)b * TT + (size_t)s * SEGLEN) * CC + c;

    float P = 1.0f, G = 0.0f;
    for (int t = 0; t < SEGLEN; ++t) {
        const size_t idx = base + (size_t)t * CC;
        const float a = at[idx];
        G = fmaf(a, G, gx[idx]);
        P *= a;
    }
    segP[(size_t)s * (BB * CC) + ch] = P;
    segG[(size_t)s * (BB * CC) + ch] = G;
}

// Pass 2 — scan the 32 carries per channel, emit h_in per segment.
__global__ __launch_bounds__(256)
void rg_lru_scan_carry(const float* __restrict__ segP, const float* __restrict__ segG,
                       float* __restrict__ segHin)
{
    const int ch = blockIdx.x * blockDim.x + threadIdx.x;    // 0 .. B*C-1
    float h = 0.0f;
    for (int s = 0; s < SEG; ++s) {
        const size_t i = (size_t)s * (BB * CC) + ch;
        segHin[i] = h;
        h = fmaf(segP[i], h, segG[i]);
    }
}

// Pass 3 — replay each segment from its correct h_in, write outputs.
__global__ __launch_bounds__(256)
void rg_lru_scan_fix(const float* __restrict__ at, const float* __restrict__ gx,
                     const float* __restrict__ segHin, float* __restrict__ out)
{
    const int tid = blockIdx.x * blockDim.x + threadIdx.x;
    const int s   = tid / (BB * CC);
    const int ch  = tid % (BB * CC);
    const int b   = ch / CC;
    const int c   = ch % CC;
    const size_t base = ((size_t)b * TT + (size_t)s * SEGLEN) * CC + c;

    float h = segHin[(size_t)s * (BB * CC) + ch];
    for (int t = 0; t < SEGLEN; ++t) {
        const size_t idx = base + (size_t)t * CC;
        h = fmaf(at[idx], h, gx[idx]);
        out[idx] = h;
    }
}

// ---------------------------------------------------------------------------
extern "C" void kernel_launch(void* const* d_in, const int* in_sizes, int n_in,
                              void* d_out, int out_size, void* d_ws, size_t ws_size,
                              hipStream_t stream)
{
    const float* x   = (const float*)d_in[0];
    const float* Wi  = (const float*)d_in[1];
    const float* bi  = (const float*)d_in[2];
    const float* Wr  = (const float*)d_in[3];
    const float* br  = (const float*)d_in[4];
    const float* lam = (const float*)d_in[5];
    float* out = (float*)d_out;

    // Workspace layout (fp32): at [N*C] | gx [N*C] | segP | segG | segHin
    float* at_ws  = (float*)d_ws;
    float* gx_ws  = at_ws + (size_t)NN * CC;
    float* segP   = gx_ws + (size_t)NN * CC;
    float* segG   = segP + (size_t)BB * CC * SEG;
    float* segHin = segG + (size_t)BB * CC * SEG;

    dim3 grid(NN / BLK_M, CC / BLK_N);   // 256 x 8 tiles
    rg_lru_gates_wmma<<<grid, 256, 0, stream>>>(x, Wi, bi, Wr, br, lam, at_ws, gx_ws);

    const int nseg_threads = BB * CC * SEG;     // 131072
    rg_lru_scan_seg  <<<nseg_threads / 256, 256, 0, stream>>>(at_ws, gx_ws, segP, segG);
    rg_lru_scan_carry<<<(BB * CC) / 256, 256, 0, stream>>>(segP, segG, segHin);
    rg_lru_scan_fix  <<<nseg_threads / 256, 256, 0, stream>>>(at_ws, gx_ws, segHin, out);
}